// IterNorm_63874753626715
// MI455X (gfx1250) — compile-verified
//
#include <hip/hip_runtime.h>
#include <cstddef>

typedef float v2f __attribute__((ext_vector_type(2)));
typedef float v8f __attribute__((ext_vector_type(8)));

#define G      4
#define NC     64
#define C_TOT  256
#define B_N    64
#define HW     3136
#define M_TOT  (B_N * HW)    // 200704
#define EPS_F  1e-5f

// ------------------------------------------------------------------
// Kernel 1: per-(g,b) partial Gram (64x64) + per-channel sums.
// 512 threads = 16 waves; wave w owns Gram tile (p,q) = (w>>2, w&3).
// ------------------------------------------------------------------
#define CHUNK  196          // 3136 = 16 * 196, multiple of 4 (K)
#define PITCH1 198          // +2 pad: conflict-free b64 LDS reads

__global__ __launch_bounds__(512)
void k_gram(const float* __restrict__ X, float* __restrict__ gram_part,
            float* __restrict__ sum_part)
{
  const int b = blockIdx.x, g = blockIdx.y;
  __shared__ float tile[64 * PITCH1];          // 50,688 B
  const int tid  = threadIdx.x;
  const int row  = tid >> 3, l8 = tid & 7;     // staging: 8 threads per row
  const int wave = tid >> 5, lane = tid & 31;
  const int mrow = lane & 15, half = lane >> 4;
  const int p = wave >> 2, q = wave & 3;

  const float* Xrow = X + (((size_t)b * C_TOT) + (size_t)g * NC + row) * HW;
  float psum = 0.f;
  v8f acc = {};

  for (int ci = 0; ci < 16; ++ci) {
    const int s0 = ci * CHUNK;
    // stage 64 x 196 floats (float4, 128B segments per 8-thread row group)
    for (int c4 = l8; c4 < CHUNK / 4; c4 += 8) {
      const float4 v = *reinterpret_cast<const float4*>(Xrow + s0 + 4 * c4);
      psum += (v.x + v.y) + (v.z + v.w);
      float* dst = &tile[row * PITCH1 + 4 * c4];
      reinterpret_cast<float2*>(dst)[0] = make_float2(v.x, v.y);
      reinterpret_cast<float2*>(dst)[1] = make_float2(v.z, v.w);
    }
    __syncthreads();
    // Gram tile accumulate: A = x rows p*16.., B = x rows q*16.. (B = A^T form)
    const float* arow = &tile[(p * 16 + mrow) * PITCH1 + 2 * half];
    const float* brow = &tile[(q * 16 + mrow) * PITCH1 + 2 * half];
    for (int kb = 0; kb < CHUNK; kb += 4) {
      v2f a  = *reinterpret_cast<const v2f*>(arow + kb);
      v2f bb = *reinterpret_cast<const v2f*>(brow + kb);
      acc = __builtin_amdgcn_wmma_f32_16x16x4_f32(false, a, false, bb,
                                                  (short)0, acc, false, false);
    }
    __syncthreads();
  }

  // write partial Gram tile (C layout: VGPR r -> rows r / r+8, col = lane%16)
  float* gp = gram_part + ((size_t)g * B_N + b) * (NC * NC);
  for (int r = 0; r < 8; ++r) {
    const int d = p * 16 + r + 8 * half;
    gp[d * NC + q * 16 + mrow] = acc[r];
  }
  // per-channel sums: reduce 8 threads of a row (deterministic, no atomics)
  for (int off = 4; off; off >>= 1) psum += __shfl_down(psum, off, 8);
  if (l8 == 0) sum_part[((size_t)g * B_N + b) * NC + row] = psum;
}

// ------------------------------------------------------------------
// Kernel 2: reduce partials -> mean, Sigma = eps*I + Gram/m - mu mu^T
// ------------------------------------------------------------------
__global__ __launch_bounds__(256)
void k_reduce(const float* __restrict__ gram_part,
              const float* __restrict__ sum_part,
              float* __restrict__ sigma, float* __restrict__ meanw)
{
  const int g = blockIdx.x;
  __shared__ float meanS[NC];
  const int tid = threadIdx.x;
  const float inv_m = 1.0f / (float)M_TOT;
  if (tid < NC) {
    float s = 0.f;
    for (int b = 0; b < B_N; ++b) s += sum_part[((size_t)g * B_N + b) * NC + tid];
    const float mu = s * inv_m;
    meanS[tid] = mu;
    meanw[g * NC + tid] = mu;
  }
  __syncthreads();
  for (int idx = tid; idx < NC * NC; idx += 256) {
    float s = 0.f;
    for (int b = 0; b < B_N; ++b)
      s += gram_part[((size_t)g * B_N + b) * (NC * NC) + idx];
    const int d = idx >> 6, e = idx & 63;
    float v = s * inv_m - meanS[d] * meanS[e];
    if (d == e) v += EPS_F;
    sigma[(size_t)g * (NC * NC) + idx] = v;
  }
}

// ------------------------------------------------------------------
// Kernel 3: Newton-Schulz (per group, one block, WMMA 64x64 matmuls)
// ------------------------------------------------------------------
#define P3P 68   // LDS pitch, conflict-free A reads

__device__ __forceinline__ void mm_lds(float* Cd, const float* A, const float* Bm,
                                       int p, int q, int mrow, int half)
{
  v8f acc = {};
  const float* ar = &A[(p * 16 + mrow) * P3P + 2 * half];
  for (int kb = 0; kb < 64; kb += 4) {
    const int col = kb + 2 * half;
    v2f a; a.x = ar[kb]; a.y = ar[kb + 1];
    v2f b; b.x = Bm[col * P3P + q * 16 + mrow];
    b.y = Bm[(col + 1) * P3P + q * 16 + mrow];
    acc = __builtin_amdgcn_wmma_f32_16x16x4_f32(false, a, false, b,
                                                (short)0, acc, false, false);
  }
  for (int r = 0; r < 8; ++r)
    Cd[(p * 16 + r + 8 * half) * P3P + q * 16 + mrow] = acc[r];
}

__global__ __launch_bounds__(512)
void k_newton(const float* __restrict__ sigma, const float* __restrict__ meanw,
              const float* __restrict__ weight, const float* __restrict__ bias,
              float* __restrict__ wmw, float* __restrict__ betaw)
{
  const int g = blockIdx.x;
  __shared__ float Pm[64 * P3P], Am[64 * P3P], Bm[64 * P3P];   // 52,224 B
  __shared__ float meanS[NC];
  __shared__ float rTrS;
  const int tid = threadIdx.x;
  const int wave = tid >> 5, lane = tid & 31;
  const int mrow = lane & 15, half = lane >> 4;
  const int p = wave >> 2, q = wave & 3;
  const float* Sig = sigma + (size_t)g * (NC * NC);

  if (tid == 0) {
    float t = 0.f;
    for (int i = 0; i < NC; ++i) t += Sig[i * NC + i];
    rTrS = 1.f / t;
  }
  if (tid < NC) meanS[tid] = meanw[g * NC + tid];
  for (int idx = tid; idx < NC * NC; idx += 512) {
    const int d = idx >> 6, e = idx & 63;
    Pm[d * P3P + e] = (d == e) ? 1.f : 0.f;
  }
  __syncthreads();
  const float rTr = rTrS;

  for (int it = 0; it < 5; ++it) {
    mm_lds(Am, Pm, Pm, p, q, mrow, half);   // A = P^2
    __syncthreads();
    mm_lds(Bm, Am, Pm, p, q, mrow, half);   // B = P^3
    __syncthreads();
    { // A = P^3 * (Sigma * rTr), Sigma streamed from L2
      v8f acc = {};
      const float* ar = &Bm[(p * 16 + mrow) * P3P + 2 * half];
      for (int kb = 0; kb < 64; kb += 4) {
        const int col = kb + 2 * half;
        v2f a; a.x = ar[kb]; a.y = ar[kb + 1];
        v2f b; b.x = Sig[col * NC + q * 16 + mrow] * rTr;
        b.y = Sig[(col + 1) * NC + q * 16 + mrow] * rTr;
        acc = __builtin_amdgcn_wmma_f32_16x16x4_f32(false, a, false, b,
                                                    (short)0, acc, false, false);
      }
      for (int r = 0; r < 8; ++r)
        Am[(p * 16 + r + 8 * half) * P3P + q * 16 + mrow] = acc[r];
    }
    __syncthreads();
    for (int idx = tid; idx < NC * NC; idx += 512) {
      const int off = (idx >> 6) * P3P + (idx & 63);
      Pm[off] = 1.5f * Pm[off] - 0.5f * Am[off];
    }
    __syncthreads();
  }

  const float s = sqrtf(rTr);
  for (int idx = tid; idx < NC * NC; idx += 512)
    wmw[(size_t)g * (NC * NC) + idx] = Pm[(idx >> 6) * P3P + (idx & 63)] * s;
  if (tid < NC) {   // beta[c] = bias - (wm . mu) * weight  (folds mean subtraction)
    float off = 0.f;
    for (int e = 0; e < NC; ++e) off += Pm[tid * P3P + e] * meanS[e];
    off *= s;
    const float w = weight[g * NC + tid];
    betaw[g * NC + tid] = bias[g * NC + tid] - off * w;
  }
}

// ------------------------------------------------------------------
// Kernel 4: Y = (wm . x) * weight + beta   (256 threads = 8 waves)
// Each wave: one 16-col i-block, all 64 output channels (4 acc tiles).
// ------------------------------------------------------------------
__global__ __launch_bounds__(256)
void k_apply(const float* __restrict__ X, const float* __restrict__ wmw,
             const float* __restrict__ weight, const float* __restrict__ betaw,
             float* __restrict__ out)
{
  const int b = blockIdx.x, g = blockIdx.y;
  __shared__ float wmS[64 * P3P];
  __shared__ float wS[NC], bS[NC];
  const int tid = threadIdx.x;
  for (int idx = tid; idx < NC * NC; idx += 256)
    wmS[(idx >> 6) * P3P + (idx & 63)] = wmw[(size_t)g * (NC * NC) + idx];
  if (tid < NC) { wS[tid] = weight[g * NC + tid]; bS[tid] = betaw[g * NC + tid]; }
  __syncthreads();

  const int wave = tid >> 5, lane = tid & 31;
  const int mrow = lane & 15, half = lane >> 4;
  float wv[4][8], bv[4][8];
  for (int p = 0; p < 4; ++p)
    for (int r = 0; r < 8; ++r) {
      const int d = p * 16 + r + 8 * half;
      wv[p][r] = wS[d]; bv[p][r] = bS[d];
    }

  const float* Xg = X   + (((size_t)b * C_TOT) + (size_t)g * NC) * HW;
  float*       Og = out + (((size_t)b * C_TOT) + (size_t)g * NC) * HW;

  for (int ib = wave; ib < HW / 16; ib += 8) {
    const int si = ib * 16 + mrow;
    v8f acc[4] = {{}, {}, {}, {}};
    for (int kb = 0; kb < 64; kb += 4) {
      const int e0 = kb + 2 * half;
      v2f bb;
      bb.x = Xg[(size_t)e0 * HW + si];
      bb.y = Xg[(size_t)(e0 + 1) * HW + si];
      for (int p = 0; p < 4; ++p) {
        v2f a = *reinterpret_cast<const v2f*>(&wmS[(p * 16 + mrow) * P3P + e0]);
        acc[p] = __builtin_amdgcn_wmma_f32_16x16x4_f32(false, a, false, bb,
                                                       (short)0, acc[p], false, false);
      }
    }
    for (int p = 0; p < 4; ++p)
      for (int r = 0; r < 8; ++r) {
        const int d = p * 16 + r + 8 * half;
        Og[(size_t)d * HW + si] = acc[p][r] * wv[p][r] + bv[p][r];
      }
  }
}

// ------------------------------------------------------------------
extern "C" void kernel_launch(void* const* d_in, const int* in_sizes, int n_in,
                              void* d_out, int out_size, void* d_ws, size_t ws_size,
                              hipStream_t stream)
{
  (void)in_sizes; (void)n_in; (void)out_size; (void)ws_size;
  const float* X      = (const float*)d_in[0];
  const float* weight = (const float*)d_in[1];
  const float* bias   = (const float*)d_in[2];
  float* out = (float*)d_out;

  float* ws        = (float*)d_ws;
  float* gram_part = ws;                                   // 4*64*4096 = 1,048,576 f
  float* sum_part  = gram_part + (size_t)G * B_N * NC * NC; // 16,384 f
  float* sigma     = sum_part  + (size_t)G * B_N * NC;      // 16,384 f
  float* meanw     = sigma     + (size_t)G * NC * NC;       // 256 f
  float* wmw       = meanw     + C_TOT;                     // 16,384 f
  float* betaw     = wmw       + (size_t)G * NC * NC;       // 256 f

  k_gram  <<<dim3(B_N, G), 512, 0, stream>>>(X, gram_part, sum_part);
  k_reduce<<<dim3(G),      256, 0, stream>>>(gram_part, sum_part, sigma, meanw);
  k_newton<<<dim3(G),      512, 0, stream>>>(sigma, meanw, weight, bias, wmw, betaw);
  k_apply <<<dim3(B_N, G), 256, 0, stream>>>(X, wmw, weight, betaw, out);
}